// SolarMoEGateReference_10840497455870
// MI455X (gfx1250) — compile-verified
//
#include <hip/hip_runtime.h>
#include <hip/hip_bf16.h>

typedef __attribute__((ext_vector_type(2))) float v2f;
typedef __attribute__((ext_vector_type(8))) float v8f;

#define HIDDEN_DIM 4096
#define NUM_EXPERTS 64
#define TOP_K 8
#define KCHUNK 256           // K values staged into LDS per chunk
#define TOKENS_PER_WG 64     // 4 waves * 16 tokens

// LDS: W chunk staged as, for each k-group g (4 consecutive k), 64 float4:
//   lds[(g*64 + n)*4 + c] = W[kc + 4g + c][n]
// so a lane's B operand pair (W[k+2hi][n], W[k+1+2hi][n]) is one aligned b64 read.
// Size: KCHUNK*64 floats = 16384 floats = 64KB. Reused for score transpose
// (4 waves * 16 tokens * pitch 65 = 4160 floats) in the epilogue.
#define LDS_FLOATS (KCHUNK * NUM_EXPERTS)
#define SC_PITCH 65

__global__ __launch_bounds__(128) void solar_moe_gate_kernel(
    const float* __restrict__ x,      // [T, 4096]
    const float* __restrict__ W,      // [4096, 64]
    const float* __restrict__ bias,   // [64]
    int*   __restrict__ out_idx,      // [T, 8]
    float* __restrict__ out_w)        // [T, 8]
{
    __shared__ float lds[LDS_FLOATS];

    const int tid   = threadIdx.x;
    const int wave  = tid >> 5;
    const int lane  = tid & 31;
    const int lhalf = lane & 15;
    const int hi    = lane >> 4;              // 0: lanes 0-15, 1: lanes 16-31
    const int rowBase = blockIdx.x * TOKENS_PER_WG + wave * 16;
    const float* xrow = x + (size_t)(rowBase + lhalf) * HIDDEN_DIM;

    v8f c0 = {}, c1 = {}, c2 = {}, c3 = {};

    for (int kc = 0; kc < HIDDEN_DIM; kc += KCHUNK) {
        __syncthreads();
        // ---- Stage W[kc .. kc+KCHUNK) into LDS (interleave-by-4 transpose) ----
        // KCHUNK*64/4 = 4096 float4 elements, 32 per thread, coalesced b128 loads.
        for (int i4 = tid; i4 < KCHUNK * (NUM_EXPERTS / 4); i4 += 128) {
            const int r  = i4 >> 4;              // row within chunk: 0..KCHUNK-1
            const int n4 = (i4 & 15) << 2;       // expert base: 0,4,...,60
            const float4 wv = *(const float4*)(W + (size_t)(kc + r) * NUM_EXPERTS + n4);
            float* dst = &lds[(r >> 2) * (NUM_EXPERTS * 4) + (r & 3)];
            dst[(n4 + 0) * 4] = wv.x;
            dst[(n4 + 1) * 4] = wv.y;
            dst[(n4 + 2) * 4] = wv.z;
            dst[(n4 + 3) * 4] = wv.w;
        }
        __syncthreads();

        // ---- K loop: 4 per step, V_WMMA_F32_16X16X4_F32 per 16-expert tile ----
        for (int g = 0; g < KCHUNK / 4; ++g) {
            const int k = kc + 4 * g;
            // A 16x4: lanes 0-15 -> (K0,K1); lanes 16-31 -> (K2,K3)
            const v2f a = *(const v2f*)&xrow[k + 2 * hi];
            // B 4x16: VGPR0 = K(2hi), VGPR1 = K(2hi+1), N striped over half-lanes
            const float* bp = &lds[g * (NUM_EXPERTS * 4)];
            const v2f b0 = *(const v2f*)&bp[(0  + lhalf) * 4 + 2 * hi];
            const v2f b1 = *(const v2f*)&bp[(16 + lhalf) * 4 + 2 * hi];
            const v2f b2 = *(const v2f*)&bp[(32 + lhalf) * 4 + 2 * hi];
            const v2f b3 = *(const v2f*)&bp[(48 + lhalf) * 4 + 2 * hi];
            c0 = __builtin_amdgcn_wmma_f32_16x16x4_f32(false, a, false, b0, (short)0, c0, false, false);
            c1 = __builtin_amdgcn_wmma_f32_16x16x4_f32(false, a, false, b1, (short)0, c1, false, false);
            c2 = __builtin_amdgcn_wmma_f32_16x16x4_f32(false, a, false, b2, (short)0, c2, false, false);
            c3 = __builtin_amdgcn_wmma_f32_16x16x4_f32(false, a, false, b3, (short)0, c3, false, false);
        }
    }

    // ---- Epilogue: transpose C tiles to LDS so one thread owns one token ----
    __syncthreads();   // all waves done reading W staging area
    {
        float* sc = &lds[wave * 16 * SC_PITCH];
        #pragma unroll
        for (int r = 0; r < 8; ++r) {
            const int m = r + 8 * hi;            // token row 0..15 within wave tile
            sc[m * SC_PITCH +  0 + lhalf] = c0[r];
            sc[m * SC_PITCH + 16 + lhalf] = c1[r];
            sc[m * SC_PITCH + 32 + lhalf] = c2[r];
            sc[m * SC_PITCH + 48 + lhalf] = c3[r];
        }
    }
    __syncthreads();

    // ---- Top-K: threads 0..63 each handle one token (all-register, no scratch) ----
    if (tid < TOKENS_PER_WG) {
        const float* scrow = &lds[(tid >> 4) * 16 * SC_PITCH + (tid & 15) * SC_PITCH];
        float s[NUM_EXPERTS];    // sigmoid(logit)         (statically indexed)
        float sb[NUM_EXPERTS];   // sigmoid(logit) + bias  (statically indexed)
        #pragma unroll
        for (int j = 0; j < NUM_EXPERTS; ++j) {
            const float z   = scrow[j];
            const float sig = 1.0f / (1.0f + __expf(-z));
            s[j]  = sig;
            sb[j] = sig + bias[j];
        }
        unsigned long long taken = 0ull;
        int   idxs[TOP_K];
        float vals[TOP_K];
        float sum = 0.0f;
        #pragma unroll
        for (int kk = 0; kk < TOP_K; ++kk) {
            float best  = -3.0e38f;
            float bestu = 0.0f;
            int   bi    = 0;
            #pragma unroll
            for (int j = 0; j < NUM_EXPERTS; ++j) {
                const bool t  = (taken >> j) & 1ull;
                const float v = t ? -3.0e38f : sb[j];
                if (v > best) { best = v; bi = j; bestu = s[j]; }
            }
            taken |= (1ull << bi);
            idxs[kk] = bi;
            vals[kk] = bestu;
            sum += bestu;
        }
        const float scale = 2.5f / (sum + 1e-20f);
        const size_t tok = (size_t)blockIdx.x * TOKENS_PER_WG + tid;
        #pragma unroll
        for (int kk = 0; kk < TOP_K; ++kk) {
            out_idx[tok * TOP_K + kk] = idxs[kk];
            out_w [tok * TOP_K + kk]  = vals[kk] * scale;
        }
    }
}

extern "C" void kernel_launch(void* const* d_in, const int* in_sizes, int n_in,
                              void* d_out, int out_size, void* d_ws, size_t ws_size,
                              hipStream_t stream) {
    const float* x    = (const float*)d_in[0];   // [T, 4096]
    const float* W    = (const float*)d_in[1];   // [4096, 64]
    const float* bias = (const float*)d_in[2];   // [64]

    const int T = in_sizes[0] / HIDDEN_DIM;      // 16384

    // d_out layout: [T*8] int32 indices, then [T*8] float32 weights
    int*   out_idx = (int*)d_out;
    float* out_w   = (float*)d_out + (size_t)T * TOP_K;

    const int blocks = T / TOKENS_PER_WG;        // 256
    solar_moe_gate_kernel<<<blocks, 128, 0, stream>>>(x, W, bias, out_idx, out_w);
}